// TMRB_71614284693962
// MI455X (gfx1250) — compile-verified
//
#include <hip/hip_runtime.h>
#include <hip/hip_bf16.h>
#include <math.h>

// Problem constants (from reference)
#define BB    16
#define NN    32768
#define DD    256
#define OUTD  256
#define TOPK  64
#define KTOT  (DD * TOPK)     // 16384
#define NCHUNK 8
#define CHUNK (NN / NCHUNK)   // 4096
#define DSPLIT 128            // d-columns (threads) per topk block

typedef __attribute__((ext_vector_type(2))) float v2f;
typedef __attribute__((ext_vector_type(8))) float v8f;

#define POS_BIG 3.402823466e38f
#define NEG_BIG (-3.402823466e38f)

// ---------------------------------------------------------------------------
// Kernel 1: per-chunk top-64 of |x[b,d,n] - h0[d]| along n, sorted descending.
// x[b,d,n] = tem_emb[b, n, d]  (lanes across d -> coalesced loads)
// ---------------------------------------------------------------------------
__global__ __launch_bounds__(DSPLIT) void topk_chunk_kernel(
    const float* __restrict__ tem, const float* __restrict__ hid,
    float* __restrict__ cand_val, int* __restrict__ cand_idx)
{
    __shared__ float          vbuf[TOPK * DSPLIT];   // [k][tid] -> conflict-free
    __shared__ unsigned short ibuf[TOPK * DSPLIT];   // chunk-local index fits u16

    const int blk   = blockIdx.x;          // B * NCHUNK * 2 blocks
    const int dhalf = blk & 1;
    const int c     = (blk >> 1) & (NCHUNK - 1);
    const int b     = blk >> 4;
    const int tid   = threadIdx.x;
    const int d     = dhalf * DSPLIT + tid;

    const float h = hid[d];                // init_hidden[d, 0]
    const float* base = tem + (size_t)b * NN * DD + (size_t)c * CHUNK * DD + d;

    // Fill first 64 candidates.
    float minDiff = POS_BIG;
    int   minPos  = 0;
    for (int i = 0; i < TOPK; ++i) {
        float v = base[(size_t)i * DD];
        vbuf[i * DSPLIT + tid] = v;
        ibuf[i * DSPLIT + tid] = (unsigned short)i;
        float df = fabsf(v - h);
        if (df < minDiff) { minDiff = df; minPos = i; }
    }

    // Scan rest of chunk: replace current min only when strictly larger
    // (strict '>' preserves the earlier index on ties, matching lax.top_k).
    for (int i = TOPK; i < CHUNK; ++i) {
        if ((i & 15) == 0 && (tid & 31) == 0) {
            __builtin_prefetch(base + (size_t)(i + 16) * DD, 0, 0);
        }
        float v  = base[(size_t)i * DD];
        float df = fabsf(v - h);
        if (df > minDiff) {
            vbuf[minPos * DSPLIT + tid] = v;
            ibuf[minPos * DSPLIT + tid] = (unsigned short)i;
            minDiff = POS_BIG;
            for (int k = 0; k < TOPK; ++k) {
                float dv = fabsf(vbuf[k * DSPLIT + tid] - h);
                if (dv < minDiff) { minDiff = dv; minPos = k; }
            }
        }
    }

    // Per-thread insertion sort: descending |diff|, tie -> ascending index.
    for (int i = 1; i < TOPK; ++i) {
        float          v  = vbuf[i * DSPLIT + tid];
        unsigned short ix = ibuf[i * DSPLIT + tid];
        float          df = fabsf(v - h);
        int j = i - 1;
        while (j >= 0) {
            float          vj = vbuf[j * DSPLIT + tid];
            unsigned short ij = ibuf[j * DSPLIT + tid];
            float          dj = fabsf(vj - h);
            bool shift = (dj < df) || (dj == df && ij > ix);
            if (!shift) break;
            vbuf[(j + 1) * DSPLIT + tid] = vj;
            ibuf[(j + 1) * DSPLIT + tid] = ij;
            --j;
        }
        vbuf[(j + 1) * DSPLIT + tid] = v;
        ibuf[(j + 1) * DSPLIT + tid] = ix;
    }

    // Emit candidate list (values + global indices) for the merge pass.
    size_t obase = (((size_t)b * NCHUNK + c) * DD + d) * TOPK;
    for (int k = 0; k < TOPK; ++k) {
        cand_val[obase + k] = vbuf[k * DSPLIT + tid];
        cand_idx[obase + k] = c * CHUNK + (int)ibuf[k * DSPLIT + tid];
    }
}

// ---------------------------------------------------------------------------
// Kernel 2: 8-way merge of sorted candidate lists -> global top-64 per (b,d).
// Writes flat[b][d*64 + k] (the 16 x 16384 A-matrix for the MLP GEMM).
// ---------------------------------------------------------------------------
__global__ __launch_bounds__(256) void merge_kernel(
    const float* __restrict__ cand_val, const int* __restrict__ cand_idx,
    const float* __restrict__ hid, float* __restrict__ flat)
{
    const int t = blockIdx.x * 256 + threadIdx.x;  // 0..4095
    const int b = t >> 8;
    const int d = t & (DD - 1);
    const float h = hid[d];

    int   cur[NCHUNK];
    float cv[NCHUNK];
    float cdf[NCHUNK];
    int   cix[NCHUNK];

#pragma unroll
    for (int cc = 0; cc < NCHUNK; ++cc) {
        size_t bse = (((size_t)b * NCHUNK + cc) * DD + d) * TOPK;
        cur[cc] = 0;
        cv[cc]  = cand_val[bse];
        cix[cc] = cand_idx[bse];
        cdf[cc] = fabsf(cv[cc] - h);
    }

    float* out = flat + ((size_t)b * DD + d) * TOPK;
    for (int k = 0; k < TOPK; ++k) {
        int   best = 0;
        float bd   = NEG_BIG;
        int   bix  = 0x7fffffff;
#pragma unroll
        for (int cc = 0; cc < NCHUNK; ++cc) {
            bool better = (cdf[cc] > bd) || (cdf[cc] == bd && cix[cc] < bix);
            if (cur[cc] < TOPK && better) { best = cc; bd = cdf[cc]; bix = cix[cc]; }
        }
        out[k] = cv[best];
#pragma unroll
        for (int cc = 0; cc < NCHUNK; ++cc) {
            if (cc == best) {
                cur[cc]++;
                if (cur[cc] < TOPK) {
                    size_t bse = (((size_t)b * NCHUNK + cc) * DD + d) * TOPK + cur[cc];
                    cv[cc]  = cand_val[bse];
                    cix[cc] = cand_idx[bse];
                    cdf[cc] = fabsf(cv[cc] - h);
                } else {
                    cdf[cc] = NEG_BIG;
                }
            }
        }
    }
}

// ---------------------------------------------------------------------------
// Kernel 3: ts_in = flat @ mlp_w.T + mlp_b via V_WMMA_F32_16X16X4_F32.
// M=16 (batch), N=256 (OUT), K=16384. One block per 16-wide N tile,
// 8 waves split-K (2048 each), LDS cross-wave reduce.
// ---------------------------------------------------------------------------
__global__ __launch_bounds__(256) void mlp_gemm_kernel(
    const float* __restrict__ flat, const float* __restrict__ mlp_w,
    const float* __restrict__ mlp_b, float* __restrict__ ts_in)
{
    const int nt   = blockIdx.x * 16;       // N tile base
    const int wave = threadIdx.x >> 5;      // 8 waves (wave32)
    const int lane = threadIdx.x & 31;
    const int m    = lane & 15;             // A row (also B col within tile)
    const int koff = (lane >> 4) << 1;      // 0 or 2 (16x4 f32 A layout)

    const int KPW = KTOT / 8;               // 2048 per wave
    const int k0  = wave * KPW;

    const float* Arow = flat + (size_t)m * KTOT;          // A: M = lane&15
    const float* Brow = mlp_w + (size_t)(nt + m) * KTOT;  // B[k][n] = mlp_w[n][k]

    v8f acc = {};
#pragma unroll 4
    for (int k = k0; k < k0 + KPW; k += 4) {
        const int kk = k + koff;
        v2f a, bv;
        a.x  = Arow[kk];
        a.y  = Arow[kk + 1];
        bv.x = Brow[kk];
        bv.y = Brow[kk + 1];
        // (neg_a, A, neg_b, B, c_mod, C, reuse_a, reuse_b)
        acc = __builtin_amdgcn_wmma_f32_16x16x4_f32(
            false, a, false, bv, (short)0, acc, false, false);
    }

    // Cross-wave split-K reduction in LDS.
    __shared__ float red[8][16][16];
#pragma unroll
    for (int r = 0; r < 8; ++r) {
        // C/D layout: VGPR r, lanes 0-15 -> M=r, lanes 16-31 -> M=r+8; N=lane&15
        int row = r + ((lane >> 4) << 3);
        int col = lane & 15;
        red[wave][row][col] = acc[r];
    }
    __syncthreads();

    const int e   = threadIdx.x;            // 256 threads -> one (m,n) each
    const int row = e >> 4;
    const int col = e & 15;
    float s = 0.0f;
#pragma unroll
    for (int w = 0; w < 8; ++w) s += red[w][row][col];
    s += mlp_b[nt + col];
    ts_in[row * OUTD + nt + col] = s;
}

// ---------------------------------------------------------------------------
// Kernel 4: GRU cell (16x256), two phases so r is complete before tanh gate.
// ---------------------------------------------------------------------------
__global__ __launch_bounds__(256) void gru_kernel(
    const float* __restrict__ ts_in, const float* __restrict__ hid,
    const float* __restrict__ Wr_w, const float* __restrict__ Wr_b,
    const float* __restrict__ Wz_w, const float* __restrict__ Wz_b,
    const float* __restrict__ Wt_w, const float* __restrict__ Wt_b,
    float* __restrict__ hnext)
{
    __shared__ float ts_s[BB * OUTD];
    __shared__ float h_s[DD];
    __shared__ float r_s[BB * OUTD];
    __shared__ float z_s[BB * OUTD];

    const int o = threadIdx.x;
    for (int i = o; i < BB * OUTD; i += 256) ts_s[i] = ts_in[i];
    h_s[o] = hid[o];
    __syncthreads();

    const float* wr = Wr_w + (size_t)o * (2 * OUTD);
    const float* wz = Wz_w + (size_t)o * (2 * OUTD);
    const float rb = Wr_b[o];
    const float zb = Wz_b[o];

    for (int b = 0; b < BB; ++b) {
        float accr = rb, accz = zb;
        const float* ts_row = &ts_s[b * OUTD];
        for (int j = 0; j < OUTD; ++j) {
            float cj = ts_row[j];
            accr += wr[j] * cj;
            accz += wz[j] * cj;
        }
        for (int j = 0; j < OUTD; ++j) {
            float cj = h_s[j];
            accr += wr[OUTD + j] * cj;
            accz += wz[OUTD + j] * cj;
        }
        r_s[b * OUTD + o] = 1.0f / (1.0f + expf(-accr));
        z_s[b * OUTD + o] = 1.0f / (1.0f + expf(-accz));
    }
    __syncthreads();

    const float* wt = Wt_w + (size_t)o * (2 * OUTD);
    const float tb = Wt_b[o];
    for (int b = 0; b < BB; ++b) {
        float acc = tb;
        const float* ts_row = &ts_s[b * OUTD];
        const float* r_row  = &r_s[b * OUTD];
        for (int j = 0; j < OUTD; ++j) acc += wt[j] * ts_row[j];
        for (int j = 0; j < OUTD; ++j) acc += wt[OUTD + j] * (h_s[j] * r_row[j]);
        float ht = tanhf(acc);
        float zz = z_s[b * OUTD + o];
        hnext[b * OUTD + o] = zz * ht + (1.0f - zz) * ts_row[o];
    }
}

// ---------------------------------------------------------------------------
// Kernel 5: broadcast h_next[b,o] across n -> 512 MB output, float4 stores.
// ---------------------------------------------------------------------------
__global__ __launch_bounds__(256) void bcast_kernel(
    const float* __restrict__ hnext, float* __restrict__ out)
{
    const int row = blockIdx.x;               // b*256 + o, 0..4095
    const float v = hnext[row];
    float4 v4 = make_float4(v, v, v, v);
    float4* o4 = (float4*)(out + (size_t)row * NN);
    const int n4 = NN / 4;                    // 8192
    for (int i = threadIdx.x; i < n4; i += 256) o4[i] = v4;
}

// ---------------------------------------------------------------------------
extern "C" void kernel_launch(void* const* d_in, const int* in_sizes, int n_in,
                              void* d_out, int out_size, void* d_ws, size_t ws_size,
                              hipStream_t stream)
{
    const float* tem   = (const float*)d_in[0];   // (B, N, D)
    // d_in[1] = year (unused by reference math)
    const float* hid   = (const float*)d_in[2];   // (D, 1)
    const float* mlp_w = (const float*)d_in[3];   // (OUT, TOPK*D)
    const float* mlp_b = (const float*)d_in[4];   // (OUT,)
    const float* Wr_w  = (const float*)d_in[5];
    const float* Wr_b  = (const float*)d_in[6];
    const float* Wz_w  = (const float*)d_in[7];
    const float* Wz_b  = (const float*)d_in[8];
    const float* Wt_w  = (const float*)d_in[9];
    const float* Wt_b  = (const float*)d_in[10];
    float* out = (float*)d_out;

    char* ws = (char*)d_ws;
    size_t off = 0;
    float* cand_val = (float*)(ws + off); off += (size_t)BB * NCHUNK * DD * TOPK * sizeof(float);
    int*   cand_idx = (int*)(ws + off);   off += (size_t)BB * NCHUNK * DD * TOPK * sizeof(int);
    float* flat     = (float*)(ws + off); off += (size_t)BB * DD * TOPK * sizeof(float);
    float* ts       = (float*)(ws + off); off += (size_t)BB * OUTD * sizeof(float);
    float* hnext    = (float*)(ws + off); off += (size_t)BB * OUTD * sizeof(float);

    topk_chunk_kernel<<<BB * NCHUNK * 2, DSPLIT, 0, stream>>>(tem, hid, cand_val, cand_idx);
    merge_kernel<<<(BB * DD) / 256, 256, 0, stream>>>(cand_val, cand_idx, hid, flat);
    mlp_gemm_kernel<<<OUTD / 16, 256, 0, stream>>>(flat, mlp_w, mlp_b, ts);
    gru_kernel<<<1, 256, 0, stream>>>(ts, hid, Wr_w, Wr_b, Wz_w, Wz_b, Wt_w, Wt_b, hnext);
    bcast_kernel<<<BB * OUTD, 256, 0, stream>>>(hnext, out);
}